// inv_loss_73358041415733
// MI455X (gfx1250) — compile-verified
//
#include <hip/hip_runtime.h>
#include <math.h>
#include <stdint.h>

// ---------------------------------------------------------------------------
// 2-D FFT magnitude-L1 loss via WMMA GEMMs on gfx1250 (MI455X).
//   Y = F X F,  F = C + iS (512x512 DFT matrix, symmetric)
//   loss = mean_s( sum |Y_s| )
// One workgroup = (slice, 16-wide column block). T[:,J] lives in LDS, so the
// 402 MB complex intermediate never touches HBM. Input read once: ~8.6 us at
// 23.3 TB/s; 309 GFLOP of f32 WMMA is cheaper than that -> bandwidth bound.
// Round-2 changes: 2-row-tile register blocking (shared B operands, 2x fewer
// ds reads per wmma, 8 independent accum chains) + async global->LDS staging
// of the F panels (GLOBAL_LOAD_ASYNC_TO_LDS_B128, ASYNCcnt-tracked).
// ---------------------------------------------------------------------------

typedef float v2f __attribute__((ext_vector_type(2)));
typedef float v8f __attribute__((ext_vector_type(8)));

#define HH 512
#define WW 512
#define NSLICE 192
#define NC 16                 // column-block width (one WMMA tile wide)
#define NROWTILES 32          // 512/16
#define LDS_FLOATS (HH * NC)  // 8192 floats = 32 KB per plane

// f32 WMMA: D(16x16,f32) = A(16x4,f32) * B(4x16,f32) + C
static __device__ __forceinline__ v8f wmma_f32(v2f a, v2f b, v8f c) {
  return __builtin_amdgcn_wmma_f32_16x16x4_f32(
      /*neg_a=*/false, a, /*neg_b=*/false, b,
      /*c_mod=*/(short)0, c, /*reuse_a=*/false, /*reuse_b=*/false);
}

// Async global -> LDS copy, 16 bytes per lane, tracked by ASYNCcnt.
static __device__ __forceinline__ void async_load_b128(uint32_t lds_byte_addr,
                                                       uint32_t glb_byte_off,
                                                       const float* base) {
  asm volatile("global_load_async_to_lds_b128 %0, %1, %2"
               :: "v"(lds_byte_addr), "v"(glb_byte_off), "s"(base)
               : "memory");
}

#if defined(__has_builtin) && __has_builtin(__builtin_amdgcn_s_wait_asynccnt)
#define WAIT_ASYNC0() __builtin_amdgcn_s_wait_asynccnt(0)
#else
#define WAIT_ASYNC0() asm volatile("s_wait_asynccnt 0x0" ::: "memory")
#endif

// ---------------------------------------------------------------------------
// Kernel 0: build DFT tables C/S in workspace, zero the loss accumulator.
// theta = -2*pi*((r*c) mod 512)/512  (exact angle reduction, f32 sincos)
// ---------------------------------------------------------------------------
__global__ void build_tables_kernel(float* __restrict__ Fc,
                                    float* __restrict__ Fs,
                                    float* __restrict__ accum) {
  int idx = blockIdx.x * blockDim.x + threadIdx.x;
  if (idx == 0) *accum = 0.0f;
  if (idx < HH * WW) {
    int r = idx >> 9;
    int c = idx & 511;
    int p = (r * c) & 511;
    float theta = -6.2831853071795864769f * (float)p * (1.0f / 512.0f);
    float s, co;
    __sincosf(theta, &s, &co);
    Fc[idx] = co;
    Fs[idx] = s;
  }
}

// ---------------------------------------------------------------------------
// Kernel 1: fused  T = X*F[:,J]  (LDS)  ->  Y[:,J] = F*T  -> sum|Y| -> atomic
// 256 threads = 8 waves; each wave owns 4 row-tiles, processed in pairs.
// ---------------------------------------------------------------------------
__launch_bounds__(256, 2)
__global__ void fft_mag_sum_kernel(const float* __restrict__ X,
                                   const float* __restrict__ Fc,
                                   const float* __restrict__ Fs,
                                   float* __restrict__ accum) {
  extern __shared__ float smem[];
  float* sFc  = smem;                   // F cos columns [512][16]   32 KB
  float* sFs  = smem + 1 * LDS_FLOATS;  // F sin columns [512][16]   32 KB
  float* sTre = smem + 2 * LDS_FLOATS;  // T real        [512][16]   32 KB
  float* sTim = smem + 3 * LDS_FLOATS;  // T imag        [512][16]   32 KB

  const int jb    = blockIdx.x;         // column block 0..31
  const int slice = blockIdx.y;         // 0..191
  const int J     = jb * NC;
  const int tid   = (int)threadIdx.x;
  const int lane  = tid & 31;
  const int wave  = tid >> 5;

  const float* __restrict__ Xs = X + (size_t)slice * (HH * WW);

  // ---- Phase 0: async-stage F[:, J:J+16] panels into LDS (b128 per lane,
  // memory->LDS direct, no VGPR round trip; ASYNCcnt-tracked).
  {
    const uint32_t ldsFc = (uint32_t)(uintptr_t)sFc;  // flat addr[31:0] = LDS offset
    const uint32_t ldsFs = (uint32_t)(uintptr_t)sFs;
    for (int q = tid; q < LDS_FLOATS / 4; q += 256) {  // 8 iters/thread
      int r  = q >> 2;
      int c4 = (q & 3) << 2;
      uint32_t goff = (uint32_t)((r * WW + J + c4) * sizeof(float));
      uint32_t loff = (uint32_t)(q * 16);
      async_load_b128(ldsFc + loff, goff, Fc);
      async_load_b128(ldsFs + loff, goff, Fs);
    }
    WAIT_ASYNC0();
  }
  __syncthreads();

  // WMMA per-lane indices (ISA 7.12.2):
  //   A 16x4 : lane m=L%16 rows; lanes<16 hold K={0,1}, lanes>=16 hold K={2,3}
  //   B 4x16 : lane n=L%16 cols; same K split per lane-half
  //   C/D    : vgpr j holds D[j + 8*(L/16)][L%16]
  const int m    = lane & 15;
  const int hi   = lane >> 4;
  const int koff = hi << 1;  // 0 or 2

  // ---- Phase 1: T[:,J] = X * F[:,J].  Two row-tiles per pass share B.
  for (int base = 0; base < NROWTILES; base += 16) {
    const int rt0 = base + wave;
    const int rt1 = rt0 + 8;
    v8f re0 = {}, im0 = {}, re1 = {}, im1 = {};
    const float* __restrict__ xrow0 = Xs + (size_t)(rt0 * 16 + m) * WW;
    const float* __restrict__ xrow1 = Xs + (size_t)(rt1 * 16 + m) * WW;
#pragma unroll 2
    for (int kb = 0; kb < 128; ++kb) {
      const int k = (kb << 2) + koff;
      v2f a0 = *(const v2f*)(xrow0 + k);            // global_load_b64
      v2f a1 = *(const v2f*)(xrow1 + k);
      v2f bc, bs;
      bc.x = sFc[k * NC + m];                       // shared across both tiles
      bc.y = sFc[(k + 1) * NC + m];
      bs.x = sFs[k * NC + m];
      bs.y = sFs[(k + 1) * NC + m];
      re0 = wmma_f32(a0, bc, re0);
      im0 = wmma_f32(a0, bs, im0);
      re1 = wmma_f32(a1, bc, re1);
      im1 = wmma_f32(a1, bs, im1);
    }
    // scatter D tiles into LDS T planes
#pragma unroll
    for (int j = 0; j < 8; ++j) {
      int row0 = rt0 * 16 + j + 8 * hi;
      int row1 = rt1 * 16 + j + 8 * hi;
      sTre[row0 * NC + m] = re0[j];
      sTim[row0 * NC + m] = im0[j];
      sTre[row1 * NC + m] = re1[j];
      sTim[row1 * NC + m] = im1[j];
    }
  }
  __syncthreads();

  // ---- Phase 2: Y[:,J] = (C + iS) * (Tre + iTim); accumulate sum |Y|.
  // Two row-tiles per pass share the B (T-panel) reads; 8 accum chains.
  float lsum = 0.0f;
  for (int base = 0; base < NROWTILES; base += 16) {
    const int rt0 = base + wave;
    const int rt1 = rt0 + 8;
    v8f cr0 = {}, si0 = {}, ci0 = {}, sr0 = {};
    v8f cr1 = {}, si1 = {}, ci1 = {}, sr1 = {};
    const float* __restrict__ fc0 = Fc + (size_t)(rt0 * 16 + m) * WW;
    const float* __restrict__ fs0 = Fs + (size_t)(rt0 * 16 + m) * WW;
    const float* __restrict__ fc1 = Fc + (size_t)(rt1 * 16 + m) * WW;
    const float* __restrict__ fs1 = Fs + (size_t)(rt1 * 16 + m) * WW;
#pragma unroll 2
    for (int kb = 0; kb < 128; ++kb) {
      const int k = (kb << 2) + koff;
      v2f ac0 = *(const v2f*)(fc0 + k);             // L2-resident table rows
      v2f as0 = *(const v2f*)(fs0 + k);
      v2f ac1 = *(const v2f*)(fc1 + k);
      v2f as1 = *(const v2f*)(fs1 + k);
      v2f bre, bim;
      bre.x = sTre[k * NC + m];
      bre.y = sTre[(k + 1) * NC + m];
      bim.x = sTim[k * NC + m];
      bim.y = sTim[(k + 1) * NC + m];
      cr0 = wmma_f32(ac0, bre, cr0);
      si0 = wmma_f32(as0, bim, si0);
      ci0 = wmma_f32(ac0, bim, ci0);
      sr0 = wmma_f32(as0, bre, sr0);
      cr1 = wmma_f32(ac1, bre, cr1);
      si1 = wmma_f32(as1, bim, si1);
      ci1 = wmma_f32(ac1, bim, ci1);
      sr1 = wmma_f32(as1, bre, sr1);
    }
#pragma unroll
    for (int j = 0; j < 8; ++j) {
      float yre0 = cr0[j] - si0[j];
      float yim0 = ci0[j] + sr0[j];
      float yre1 = cr1[j] - si1[j];
      float yim1 = ci1[j] + sr1[j];
      lsum += sqrtf(yre0 * yre0 + yim0 * yim0);
      lsum += sqrtf(yre1 * yre1 + yim1 * yim1);
    }
  }

  // wave32 butterfly reduction, then one atomic per wave
#pragma unroll
  for (int off = 16; off > 0; off >>= 1)
    lsum += __shfl_xor(lsum, off, 32);
  if (lane == 0) atomicAdd(accum, lsum);
}

// ---------------------------------------------------------------------------
// Kernel 2: scale by 1/192 (mean over 64*3 slices, LOSS_WEIGHT = 1)
// ---------------------------------------------------------------------------
__global__ void finalize_kernel(const float* __restrict__ accum,
                                float* __restrict__ out) {
  if (threadIdx.x == 0 && blockIdx.x == 0)
    out[0] = accum[0] * (1.0f / (float)NSLICE);
}

// ---------------------------------------------------------------------------
extern "C" void kernel_launch(void* const* d_in, const int* in_sizes, int n_in,
                              void* d_out, int out_size, void* d_ws, size_t ws_size,
                              hipStream_t stream) {
  const float* k_in = (const float*)d_in[0];   // [64,3,512,512] f32
  float* out = (float*)d_out;                  // scalar f32

  float* Fc    = (float*)d_ws;                 // 1 MB
  float* Fs    = Fc + HH * WW;                 // 1 MB
  float* accum = Fs + HH * WW;                 // 4 B

  build_tables_kernel<<<(HH * WW + 255) / 256, 256, 0, stream>>>(Fc, Fs, accum);

  dim3 grid(WW / NC, NSLICE);                  // 32 x 192 workgroups
  size_t lds_bytes = 4 * LDS_FLOATS * sizeof(float);  // 128 KB dynamic LDS
  fft_mag_sum_kernel<<<grid, 256, lds_bytes, stream>>>(k_in, Fc, Fs, accum);

  finalize_kernel<<<1, 1, 0, stream>>>(accum, out);
}